// GraphSage_58420145160634
// MI455X (gfx1250) — compile-verified
//
#include <hip/hip_runtime.h>

typedef float v2f __attribute__((ext_vector_type(2)));
typedef float v8f __attribute__((ext_vector_type(8)));

#define N_NODES_C 50000
#define DEG_C     16
#define FEAT_C    128
#define HID_C     128
#define OUT_C     64
#define NPB       64      // nodes per block
#define THREADS   256     // 8 waves (wave32)

// ---------------------------------------------------------------------------
// WMMA helper: D = A(16x4 f32) * B(4x16 f32) + C, via V_WMMA_F32_16X16X4_F32
// A frag: lane<16 holds A[m][0..1], lane>=16 holds A[m][2..3]  (m = lane&15)
// B frag: lane<16 holds B[0..1][n], lane>=16 holds B[2..3][n]  (n = lane&15)
// D frag: vgpr r holds D[r + (lane<16?0:8)][lane&15]
// ---------------------------------------------------------------------------
__device__ __forceinline__ v8f wmma_f32_k4(v2f a, v2f b, v8f c) {
    return __builtin_amdgcn_wmma_f32_16x16x4_f32(
        false, a, false, b, (short)0, c, false, false);
}

// ===========================================================================
// Kernel 1: h1[v] = relu( concat(x_v, mean_{u in N(v)} x_u) @ w1^T )
// ===========================================================================
__global__ __launch_bounds__(THREADS)
void sage_layer1(const float* __restrict__ feat,
                 const float* __restrict__ w1,     // [128][256] row-major
                 const int*   __restrict__ nidx,   // [N][16]
                 float*       __restrict__ h1)     // [N][128]
{
    __shared__ float xcat[NPB * 256];   // 64 KB: [node][ x(128) | mean(128) ]

    const int tid   = threadIdx.x;
    const int node0 = blockIdx.x * NPB;

    // ---- Phase 1: gather features + neighbor mean into LDS ----------------
    {
        const int m    = tid >> 2;          // 0..63 node within block
        const int part = tid & 3;           // 4 threads/node, 32 cols each
        int nid = node0 + m;
        if (nid >= N_NODES_C) nid = N_NODES_C - 1;

        const float4* self = (const float4*)(feat + (size_t)nid * FEAT_C) + part * 8;
        float4* dstL = (float4*)(xcat + m * 256) + part * 8;
        float4* dstR = (float4*)(xcat + m * 256 + FEAT_C) + part * 8;

        float4 acc[8];
        #pragma unroll
        for (int c = 0; c < 8; ++c) acc[c] = make_float4(0.f, 0.f, 0.f, 0.f);

        #pragma unroll 4
        for (int j = 0; j < DEG_C; ++j) {
            const int nb = nidx[(size_t)nid * DEG_C + j];
            const float4* row = (const float4*)(feat + (size_t)nb * FEAT_C) + part * 8;
            #pragma unroll
            for (int c = 0; c < 8; ++c) {
                float4 v = row[c];
                acc[c].x += v.x; acc[c].y += v.y;
                acc[c].z += v.z; acc[c].w += v.w;
            }
        }
        const float inv = 1.0f / (float)DEG_C;
        #pragma unroll
        for (int c = 0; c < 8; ++c) {
            dstL[c] = self[c];
            float4 a = acc[c];
            dstR[c] = make_float4(a.x * inv, a.y * inv, a.z * inv, a.w * inv);
        }
    }
    __syncthreads();

    // ---- Phase 2: WMMA GEMM: [64x256] @ [256x128] + relu ------------------
    const int wave   = tid >> 5;
    const int lane   = tid & 31;
    const int mt     = wave >> 1;            // 0..3 : 16-row M tile
    const int ntBase = (wave & 1) * 4;       // 0 or 4 : four 16-col N tiles
    const int mrow   = lane & 15;
    const int koff   = (lane < 16) ? 0 : 2;

    v8f c0 = {}, c1 = {}, c2 = {}, c3 = {};
    const float* aBase  = xcat + (mt * 16 + mrow) * 256 + koff;
    const float* bBase0 = w1 + (size_t)((ntBase + 0) * 16 + mrow) * 256 + koff;
    const float* bBase1 = w1 + (size_t)((ntBase + 1) * 16 + mrow) * 256 + koff;
    const float* bBase2 = w1 + (size_t)((ntBase + 2) * 16 + mrow) * 256 + koff;
    const float* bBase3 = w1 + (size_t)((ntBase + 3) * 16 + mrow) * 256 + koff;

    #pragma unroll 8
    for (int kk = 0; kk < 64; ++kk) {
        const int k0 = kk * 4;
        v2f a  = *(const v2f*)(aBase  + k0);
        v2f b0 = *(const v2f*)(bBase0 + k0);
        v2f b1 = *(const v2f*)(bBase1 + k0);
        v2f b2 = *(const v2f*)(bBase2 + k0);
        v2f b3 = *(const v2f*)(bBase3 + k0);
        c0 = wmma_f32_k4(a, b0, c0);
        c1 = wmma_f32_k4(a, b1, c1);
        c2 = wmma_f32_k4(a, b2, c2);
        c3 = wmma_f32_k4(a, b3, c3);
    }

    // ---- relu + store h1 ---------------------------------------------------
    const int rbase = (lane < 16) ? 0 : 8;
    #pragma unroll
    for (int r = 0; r < 8; ++r) {
        const int g = node0 + mt * 16 + rbase + r;
        if (g < N_NODES_C) {
            float* o = h1 + (size_t)g * HID_C;
            float v;
            v = c0[r]; o[(ntBase + 0) * 16 + mrow] = v > 0.f ? v : 0.f;
            v = c1[r]; o[(ntBase + 1) * 16 + mrow] = v > 0.f ? v : 0.f;
            v = c2[r]; o[(ntBase + 2) * 16 + mrow] = v > 0.f ? v : 0.f;
            v = c3[r]; o[(ntBase + 3) * 16 + mrow] = v > 0.f ? v : 0.f;
        }
    }
}

// ===========================================================================
// Kernel 2: h2 = relu( concat(h1_v, mean h1_nbr) @ w2^T ) ; out = h2 @ cw
// ===========================================================================
__global__ __launch_bounds__(THREADS)
void sage_layer2(const float* __restrict__ h1,     // [N][128]
                 const float* __restrict__ w2,     // [128][256]
                 const float* __restrict__ cw,     // [128][64]
                 const int*   __restrict__ nidx,   // [N][16]
                 const int*   __restrict__ nodes,  // [N]
                 float*       __restrict__ outp)   // [N][64]
{
    __shared__ float xcat[NPB * 256];   // 64 KB
    __shared__ float h2s [NPB * 128];   // 32 KB

    const int tid   = threadIdx.x;
    const int node0 = blockIdx.x * NPB;

    // ---- Phase 1: gather h1 + neighbor mean into LDS ----------------------
    {
        const int m    = tid >> 2;
        const int part = tid & 3;
        int b = node0 + m;
        if (b >= N_NODES_C) b = N_NODES_C - 1;
        const int nid = nodes[b];

        const float4* self = (const float4*)(h1 + (size_t)nid * HID_C) + part * 8;
        float4* dstL = (float4*)(xcat + m * 256) + part * 8;
        float4* dstR = (float4*)(xcat + m * 256 + HID_C) + part * 8;

        float4 acc[8];
        #pragma unroll
        for (int c = 0; c < 8; ++c) acc[c] = make_float4(0.f, 0.f, 0.f, 0.f);

        #pragma unroll 4
        for (int j = 0; j < DEG_C; ++j) {
            const int nb = nidx[(size_t)nid * DEG_C + j];
            const float4* row = (const float4*)(h1 + (size_t)nb * HID_C) + part * 8;
            #pragma unroll
            for (int c = 0; c < 8; ++c) {
                float4 v = row[c];
                acc[c].x += v.x; acc[c].y += v.y;
                acc[c].z += v.z; acc[c].w += v.w;
            }
        }
        const float inv = 1.0f / (float)DEG_C;
        #pragma unroll
        for (int c = 0; c < 8; ++c) {
            dstL[c] = self[c];
            float4 a = acc[c];
            dstR[c] = make_float4(a.x * inv, a.y * inv, a.z * inv, a.w * inv);
        }
    }
    __syncthreads();

    const int wave = tid >> 5;
    const int lane = tid & 31;
    const int mrow = lane & 15;
    const int koff = (lane < 16) ? 0 : 2;

    // ---- Phase 2: GEMM2 [64x256]@[256x128] + relu -> h2s in LDS -----------
    {
        const int mt     = wave >> 1;
        const int ntBase = (wave & 1) * 4;

        v8f c0 = {}, c1 = {}, c2 = {}, c3 = {};
        const float* aBase  = xcat + (mt * 16 + mrow) * 256 + koff;
        const float* bBase0 = w2 + (size_t)((ntBase + 0) * 16 + mrow) * 256 + koff;
        const float* bBase1 = w2 + (size_t)((ntBase + 1) * 16 + mrow) * 256 + koff;
        const float* bBase2 = w2 + (size_t)((ntBase + 2) * 16 + mrow) * 256 + koff;
        const float* bBase3 = w2 + (size_t)((ntBase + 3) * 16 + mrow) * 256 + koff;

        #pragma unroll 8
        for (int kk = 0; kk < 64; ++kk) {
            const int k0 = kk * 4;
            v2f a  = *(const v2f*)(aBase  + k0);
            v2f b0 = *(const v2f*)(bBase0 + k0);
            v2f b1 = *(const v2f*)(bBase1 + k0);
            v2f b2 = *(const v2f*)(bBase2 + k0);
            v2f b3 = *(const v2f*)(bBase3 + k0);
            c0 = wmma_f32_k4(a, b0, c0);
            c1 = wmma_f32_k4(a, b1, c1);
            c2 = wmma_f32_k4(a, b2, c2);
            c3 = wmma_f32_k4(a, b3, c3);
        }

        const int rbase = (lane < 16) ? 0 : 8;
        #pragma unroll
        for (int r = 0; r < 8; ++r) {
            const int row = mt * 16 + rbase + r;
            float* o = h2s + row * 128;
            float v;
            v = c0[r]; o[(ntBase + 0) * 16 + mrow] = v > 0.f ? v : 0.f;
            v = c1[r]; o[(ntBase + 1) * 16 + mrow] = v > 0.f ? v : 0.f;
            v = c2[r]; o[(ntBase + 2) * 16 + mrow] = v > 0.f ? v : 0.f;
            v = c3[r]; o[(ntBase + 3) * 16 + mrow] = v > 0.f ? v : 0.f;
        }
    }
    __syncthreads();

    // ---- Phase 3: GEMM3 [64x128]@[128x64] -> out --------------------------
    {
        const int mt     = wave >> 1;             // 0..3
        const int ntBase = (wave & 1) * 2;        // 0 or 2 : two 16-col tiles
        const int n0 = (ntBase + 0) * 16 + mrow;
        const int n1 = (ntBase + 1) * 16 + mrow;

        v8f c0 = {}, c1 = {};
        const float* aBase = h2s + (mt * 16 + mrow) * 128 + koff;

        #pragma unroll 8
        for (int kk = 0; kk < 32; ++kk) {
            const int k0 = kk * 4;
            v2f a = *(const v2f*)(aBase + k0);
            v2f b0, b1;
            b0.x = cw[(size_t)(k0 + koff + 0) * OUT_C + n0];
            b0.y = cw[(size_t)(k0 + koff + 1) * OUT_C + n0];
            b1.x = cw[(size_t)(k0 + koff + 0) * OUT_C + n1];
            b1.y = cw[(size_t)(k0 + koff + 1) * OUT_C + n1];
            c0 = wmma_f32_k4(a, b0, c0);
            c1 = wmma_f32_k4(a, b1, c1);
        }

        const int rbase = (lane < 16) ? 0 : 8;
        #pragma unroll
        for (int r = 0; r < 8; ++r) {
            const int g = node0 + mt * 16 + rbase + r;
            if (g < N_NODES_C) {
                float* o = outp + (size_t)g * OUT_C;
                o[n0] = c0[r];
                o[n1] = c1[r];
            }
        }
    }
}

// ===========================================================================
extern "C" void kernel_launch(void* const* d_in, const int* in_sizes, int n_in,
                              void* d_out, int out_size, void* d_ws, size_t ws_size,
                              hipStream_t stream) {
    const float* feat  = (const float*)d_in[0];   // [50000*128]
    const float* w1    = (const float*)d_in[1];   // [128*256]
    const float* w2    = (const float*)d_in[2];   // [128*256]
    const float* cw    = (const float*)d_in[3];   // [128*64]
    const int*   nidx  = (const int*)d_in[4];     // [50000*16]
    const int*   nodes = (const int*)d_in[5];     // [50000]
    float* outp = (float*)d_out;                  // [50000*64]
    float* h1   = (float*)d_ws;                   // [50000*128] = 25.6 MB scratch

    const int blocks = (N_NODES_C + NPB - 1) / NPB;   // 782
    sage_layer1<<<blocks, THREADS, 0, stream>>>(feat, w1, nidx, h1);
    sage_layer2<<<blocks, THREADS, 0, stream>>>(h1, w2, cw, nidx, nodes, outp);
}